// ResGraphConvBlock_28329604284661
// MI455X (gfx1250) — compile-verified
//
#include <hip/hip_runtime.h>
#include <hip/hip_bf16.h>

typedef __attribute__((ext_vector_type(2))) float v2f;
typedef __attribute__((ext_vector_type(8))) float v8f;

#define FDIM 128
#define NUM_GRAPHS 64
#define GN_EPS 1e-5f
#define TILE_K 32

// ---------------- degree / norm ----------------
__global__ void deg_kernel(const float* __restrict__ ew, const int* __restrict__ dst,
                           float* __restrict__ deg, int E) {
    int e = blockIdx.x * blockDim.x + threadIdx.x;
    if (e < E) atomicAdd(&deg[dst[e]], ew[e]);
}

__global__ void dinv_kernel(float* __restrict__ deg, int N) {
    int i = blockIdx.x * blockDim.x + threadIdx.x;
    if (i < N) {
        float d = deg[i];
        deg[i] = (d > 0.0f) ? rsqrtf(fmaxf(d, 1e-30f)) : 0.0f;
    }
}

__global__ void norm_kernel(const int* __restrict__ src, const int* __restrict__ dst,
                            const float* __restrict__ ew, const float* __restrict__ dinv,
                            float* __restrict__ nrm, int E) {
    int e = blockIdx.x * blockDim.x + threadIdx.x;
    if (e < E) nrm[e] = dinv[src[e]] * ew[e] * dinv[dst[e]];
}

// ---------------- one hop: wave per edge, float4 per lane ----------------
__global__ __launch_bounds__(256) void propagate_kernel(
    const float* __restrict__ h, const int* __restrict__ src, const int* __restrict__ dst,
    const float* __restrict__ nrm, float* __restrict__ hnext, int E) {
    int wave = (int)((blockIdx.x * blockDim.x + threadIdx.x) >> 5);
    int lane = threadIdx.x & 31;
    if (wave >= E) return;
    int s = src[wave];
    int d = dst[wave];
    float w = nrm[wave];
    float4 v = ((const float4*)(h + (size_t)s * FDIM))[lane];
    float* hd = hnext + (size_t)d * FDIM + lane * 4;
    atomicAdd(hd + 0, w * v.x);
    atomicAdd(hd + 1, w * v.y);
    atomicAdd(hd + 2, w * v.z);
    atomicAdd(hd + 3, w * v.w);
}

// ---------------- fp32 WMMA GEMM: out[M,128] (+)= A[M,128] @ W[128,128] ----------------
// wave computes one 16-row M tile x full N=128 (8 accumulator fragments)
__global__ __launch_bounds__(128) void gemm_tag_kernel(
    const float* __restrict__ A, const float* __restrict__ W,
    const float* __restrict__ bias, float* __restrict__ out,
    int N, int Mtiles, int accumulate, int addBias) {
    __shared__ float sW[TILE_K * FDIM];  // 16 KB chunk of the weight matrix

    const int lane = threadIdx.x & 31;
    const int waveInBlk = threadIdx.x >> 5;
    int tile = blockIdx.x * 4 + waveInBlk;
    const bool active = (tile < Mtiles);
    if (!active) tile = 0;  // keep barriers convergent; loads stay in-bounds

    const int halfSel = lane >> 4;  // 0: lanes 0-15, 1: lanes 16-31
    const int l15 = lane & 15;
    int m = tile * 16 + l15;
    if (m >= N) m = N - 1;  // clamp (N=50000 is exact; safety only)

    v8f acc[8] = {};

    for (int k0 = 0; k0 < FDIM; k0 += TILE_K) {
        __syncthreads();  // protect previous chunk's readers
        // stage TILE_K rows of W into LDS (128 threads x float4)
        for (int i = threadIdx.x; i < TILE_K * FDIM / 4; i += 128) {
            ((float4*)sW)[i] = ((const float4*)(W + (size_t)k0 * FDIM))[i];
        }
        __syncthreads();

#pragma unroll
        for (int kk = 0; kk < TILE_K; kk += 4) {
            const int kb = kk + 2 * halfSel;  // K pair owned by this half-wave
            // A fragment: row m, K = k0+kb, k0+kb+1  (8-byte aligned)
            v2f a = *(const v2f*)(A + (size_t)m * FDIM + (k0 + kb));
#pragma unroll
            for (int nt = 0; nt < 8; ++nt) {
                v2f b;
                b.x = sW[kb * FDIM + nt * 16 + l15];        // rows K, K+2
                b.y = sW[(kb + 1) * FDIM + nt * 16 + l15];  // rows K+1, K+3
                acc[nt] = __builtin_amdgcn_wmma_f32_16x16x4_f32(
                    false, a, false, b, (short)0, acc[nt], false, false);
            }
        }
    }

    if (active) {
#pragma unroll
        for (int nt = 0; nt < 8; ++nt) {
            const int col = nt * 16 + l15;
#pragma unroll
            for (int r = 0; r < 8; ++r) {
                const int row = tile * 16 + r + halfSel * 8;
                if (row < N) {
                    const size_t idx = (size_t)row * FDIM + col;
                    float v = acc[nt][r];
                    if (accumulate) v += out[idx];
                    if (addBias) v += bias[col];
                    out[idx] = v;
                }
            }
        }
    }
}

// ---------------- GraphNorm ----------------
__global__ void cnt_kernel(const int* __restrict__ batch, float* __restrict__ cnt, int N) {
    int i = blockIdx.x * blockDim.x + threadIdx.x;
    if (i < N) atomicAdd(&cnt[batch[i]], 1.0f);
}

__global__ void sum_kernel(const float* __restrict__ out, const int* __restrict__ batch,
                           float* __restrict__ gsum, int N) {
    int idx = blockIdx.x * blockDim.x + threadIdx.x;
    if (idx < N * FDIM) {
        int node = idx >> 7;
        int f = idx & (FDIM - 1);
        atomicAdd(&gsum[batch[node] * FDIM + f], out[idx]);
    }
}

__global__ void div_kernel(float* __restrict__ arr, const float* __restrict__ cnt) {
    int idx = blockIdx.x * blockDim.x + threadIdx.x;
    if (idx < NUM_GRAPHS * FDIM) {
        arr[idx] /= fmaxf(cnt[idx >> 7], 1.0f);
    }
}

__global__ void var_kernel(const float* __restrict__ out, const int* __restrict__ batch,
                           const float* __restrict__ mean, const float* __restrict__ ms,
                           float* __restrict__ gvar, int N) {
    int idx = blockIdx.x * blockDim.x + threadIdx.x;
    if (idx < N * FDIM) {
        int node = idx >> 7;
        int f = idx & (FDIM - 1);
        float xc = out[idx] - ms[f] * mean[batch[node] * FDIM + f];
        atomicAdd(&gvar[batch[node] * FDIM + f], xc * xc);
    }
}

__global__ void final_kernel(const float* __restrict__ y, const float* __restrict__ out,
                             const int* __restrict__ batch, const float* __restrict__ mean,
                             const float* __restrict__ gvar, const float* __restrict__ ms,
                             const float* __restrict__ gw, const float* __restrict__ gb,
                             float* __restrict__ res, int N) {
    int idx = blockIdx.x * blockDim.x + threadIdx.x;
    if (idx < N * FDIM) {
        int node = idx >> 7;
        int f = idx & (FDIM - 1);
        int g = batch[node];
        float xc = out[idx] - ms[f] * mean[g * FDIM + f];
        float hn = gw[f] * xc * rsqrtf(gvar[g * FDIM + f] + GN_EPS) + gb[f];
        res[idx] = y[idx] + fmaxf(hn, 0.0f);
    }
}

// ---------------- launch ----------------
extern "C" void kernel_launch(void* const* d_in, const int* in_sizes, int n_in,
                              void* d_out, int out_size, void* d_ws, size_t ws_size,
                              hipStream_t stream) {
    const float* y     = (const float*)d_in[0];
    const int*   eidx  = (const int*)d_in[1];
    const float* ew    = (const float*)d_in[2];
    const int*   batch = (const int*)d_in[3];
    const float* tagw  = (const float*)d_in[4];  // [4,128,128]
    const float* tagb  = (const float*)d_in[5];  // [128]
    const float* gw    = (const float*)d_in[6];
    const float* gb    = (const float*)d_in[7];
    const float* gms   = (const float*)d_in[8];
    float* res = (float*)d_out;

    const int N = in_sizes[0] / FDIM;   // 50000
    const int E = in_sizes[2];          // 625000
    const int* src = eidx;
    const int* dst = eidx + E;

    // ---- workspace partition (256B aligned) ----
    size_t off = 0;
    auto take = [&](size_t bytes) {
        void* p = (char*)d_ws + off;
        off += (bytes + 255) & ~(size_t)255;
        return p;
    };
    float* deg  = (float*)take((size_t)N * 4);            // degree -> dinv in place
    float* nrm  = (float*)take((size_t)E * 4);            // edge norms
    float* hA   = (float*)take((size_t)N * FDIM * 4);     // hop buffer A
    float* hB   = (float*)take((size_t)N * FDIM * 4);     // hop buffer B
    float* outb = (float*)take((size_t)N * FDIM * 4);     // TAGConv output
    float* gsum = (float*)take((size_t)NUM_GRAPHS * FDIM * 4);
    float* gvar = (float*)take((size_t)NUM_GRAPHS * FDIM * 4);
    float* gcnt = (float*)take((size_t)NUM_GRAPHS * 4);

    const int tb = 256;
    const int gE  = (E + tb - 1) / tb;
    const int gN  = (N + tb - 1) / tb;
    const int gNF = (N * FDIM + tb - 1) / tb;
    const int gProp = (E * 32 + tb - 1) / tb;             // wave per edge
    const int Mtiles = (N + 15) / 16;
    const int gGemm = (Mtiles + 3) / 4;                   // 4 waves per block

    // ---- gcn_norm ----
    hipMemsetAsync(deg, 0, (size_t)N * 4, stream);
    deg_kernel<<<gE, tb, 0, stream>>>(ew, dst, deg, E);
    dinv_kernel<<<gN, tb, 0, stream>>>(deg, N);
    norm_kernel<<<gE, tb, 0, stream>>>(src, dst, ew, deg, nrm, E);

    // ---- hop 0: out = y @ W0 ----
    gemm_tag_kernel<<<gGemm, 128, 0, stream>>>(y, tagw, tagb, outb, N, Mtiles, 0, 0);

    // ---- hop 1 ----
    hipMemsetAsync(hA, 0, (size_t)N * FDIM * 4, stream);
    propagate_kernel<<<gProp, tb, 0, stream>>>(y, src, dst, nrm, hA, E);
    gemm_tag_kernel<<<gGemm, 128, 0, stream>>>(hA, tagw + 1 * FDIM * FDIM, tagb, outb, N, Mtiles, 1, 0);

    // ---- hop 2 ----
    hipMemsetAsync(hB, 0, (size_t)N * FDIM * 4, stream);
    propagate_kernel<<<gProp, tb, 0, stream>>>(hA, src, dst, nrm, hB, E);
    gemm_tag_kernel<<<gGemm, 128, 0, stream>>>(hB, tagw + 2 * FDIM * FDIM, tagb, outb, N, Mtiles, 1, 0);

    // ---- hop 3 (+ bias) ----
    hipMemsetAsync(hA, 0, (size_t)N * FDIM * 4, stream);
    propagate_kernel<<<gProp, tb, 0, stream>>>(hB, src, dst, nrm, hA, E);
    gemm_tag_kernel<<<gGemm, 128, 0, stream>>>(hA, tagw + 3 * FDIM * FDIM, tagb, outb, N, Mtiles, 1, 1);

    // ---- GraphNorm ----
    hipMemsetAsync(gsum, 0, (size_t)NUM_GRAPHS * FDIM * 4, stream);
    hipMemsetAsync(gvar, 0, (size_t)NUM_GRAPHS * FDIM * 4, stream);
    hipMemsetAsync(gcnt, 0, (size_t)NUM_GRAPHS * 4, stream);
    cnt_kernel<<<gN, tb, 0, stream>>>(batch, gcnt, N);
    sum_kernel<<<gNF, tb, 0, stream>>>(outb, batch, gsum, N);
    div_kernel<<<(NUM_GRAPHS * FDIM + tb - 1) / tb, tb, 0, stream>>>(gsum, gcnt);  // -> mean
    var_kernel<<<gNF, tb, 0, stream>>>(outb, batch, gsum, gms, gvar, N);
    div_kernel<<<(NUM_GRAPHS * FDIM + tb - 1) / tb, tb, 0, stream>>>(gvar, gcnt);  // -> var

    // ---- normalize + ReLU + residual ----
    final_kernel<<<gNF, tb, 0, stream>>>(y, outb, batch, gsum, gvar, gms, gw, gb, res, N);
}